// EdgeCycle_50869592655546
// MI455X (gfx1250) — compile-verified
//
#include <hip/hip_runtime.h>
#include <cstdint>

// ---------------------------------------------------------------------------
// EdgeCycle GNN block for MI455X (gfx1250, wave32, WMMA f32 16x16x4)
// ---------------------------------------------------------------------------

#define H_DIM 128
#define RE_ROWS 100000
#define RC_ROWS 120000
#define NE_INST 50000
#define NC_INST 20000
#define M1_CNT 600000
#define M2_CNT 300000
#define BN_EPS 1e-5f

typedef __attribute__((ext_vector_type(2))) float v2f;
typedef __attribute__((ext_vector_type(8))) float v8f;

// ---------------------------------------------------------------------------
// Zero a float region
__global__ void k_zero(float* p, long n) {
    long i = (long)blockIdx.x * blockDim.x + threadIdx.x;
    if (i < n) p[i] = 0.f;
}

// Build Xc[RC,640]: cols 0-127 = 0 (idp2), 256-383 = 0 (idp1), 512-639 = cycle_attr
__global__ void k_init_xc(float* __restrict__ Xc, const float* __restrict__ cyc) {
    long i = (long)blockIdx.x * blockDim.x + threadIdx.x;
    if (i >= (long)RC_ROWS * 128) return;
    long r = i >> 7; int c = (int)(i & 127);
    float* row = Xc + r * 640;
    row[c] = 0.f;
    row[256 + c] = 0.f;
    row[512 + c] = cyc[i];
}

// Build Xe[RE,256]: cols 0-127 = (1+eps_e1)*edge_attr (scatter target)
__global__ void k_init_xe(float* __restrict__ Xe, const float* __restrict__ ea,
                          const float* __restrict__ eps1) {
    long i = (long)blockIdx.x * blockDim.x + threadIdx.x;
    if (i >= (long)RE_ROWS * 128) return;
    long r = i >> 7; int c = (int)(i & 127);
    Xe[r * 256 + c] = (1.0f + eps1[0]) * ea[i];
}

// Scatter-add: one wave per message, lane handles 4 floats (float4 src read).
// out[dst[m]][col_off + 0..127] += scale * X[src[m]][0..127]
__global__ void k_scatter(const float* __restrict__ X,
                          const int* __restrict__ src, const int* __restrict__ dst,
                          int M, float* __restrict__ out, int ld_out, int col_off,
                          const float* __restrict__ eps) {
    int m = (int)(((long)blockIdx.x * blockDim.x + threadIdx.x) >> 5);
    int lane = threadIdx.x & 31;
    if (m >= M) return;
    float sc = eps ? (1.0f + eps[0]) : 1.0f;
    int s = src[m], d = dst[m];
    float4 v = *(const float4*)(X + (long)s * 128 + lane * 4);
    float* p = out + (long)d * ld_out + col_off + lane * 4;
    unsafeAtomicAdd(p + 0, sc * v.x);
    unsafeAtomicAdd(p + 1, sc * v.y);
    unsafeAtomicAdd(p + 2, sc * v.z);
    unsafeAtomicAdd(p + 3, sc * v.w);
}

// Per-instance sum of 128 cols at src_off, broadcast into cols at dst_off.
__global__ void k_sum_broadcast(float* __restrict__ buf, int ld, int so, int po,
                                int rpi, int n_inst) {
    long i = (long)blockIdx.x * blockDim.x + threadIdx.x;
    if (i >= (long)n_inst * 128) return;
    int c = (int)(i & 127); long inst = i >> 7;
    long base = inst * (long)rpi * ld;
    float s = 0.f;
    for (int k = 0; k < rpi; ++k) s += buf[base + (long)k * ld + so + c];
    for (int k = 0; k < rpi; ++k) buf[base + (long)k * ld + po + c] = s;
}

// Per-instance sum of x[inst*rpi + k][c] -> out[inst][c]
__global__ void k_sum_per_inst(const float* __restrict__ x, float* __restrict__ out,
                               int rpi, int n_inst) {
    long i = (long)blockIdx.x * blockDim.x + threadIdx.x;
    if (i >= (long)n_inst * 128) return;
    int c = (int)(i & 127); long inst = i >> 7;
    const float* base = x + inst * (long)rpi * 128;
    float s = 0.f;
    for (int k = 0; k < rpi; ++k) s += base[(long)k * 128 + c];
    out[inst * 128 + c] = s;
}

// ---------------------------------------------------------------------------
// WMMA fp32 GEMM: Y[N,dout] = X[N,K](lda) @ W[K,dout] + bias
// Block: 256 threads = 8 waves; wave w -> rows [blk*128 + w*16, +16),
// block covers 128 output cols (blockIdx.y selects 128-col slab).
// Software pipelined: chunk k+1 global->regs overlaps chunk k WMMAs;
// regs->LDS only at the chunk boundary. A [128][34] padded, B^T [128][34].
#define KC 32
#define LPAD 34

__global__ __launch_bounds__(256) void k_gemm(
        const float* __restrict__ X, int N, int K, int lda,
        const float* __restrict__ W, int dout,
        const float* __restrict__ bias, float* __restrict__ Y) {
    __shared__ float As[128 * LPAD];
    __shared__ float Bs[128 * LPAD];
    const int tid = threadIdx.x;
    const int lane = tid & 31;
    const int w = tid >> 5;
    const long rowBase = (long)blockIdx.x * 128;
    const int colBase = blockIdx.y * 128;
    const bool fullRows = (rowBase + 128 <= (long)N);

    v8f acc[8] = {};

    const int am = lane & 15;           // row within 16 (A) / col within 16 (B,C)
    const int ak = (lane >> 4) << 1;    // k offset 0 or 2 for this half-wave

    // Staging coordinates: A element i -> row (tid>>5)+i*8, k = tid&31
    //                      B element i -> k   (tid>>7)+i*2, col = tid&127
    const int arow = tid >> 5, akk = tid & 31;
    const int bkr0 = tid >> 7, bcc = tid & 127;

    float aReg[16], bReg[16];

    auto loadChunk = [&](int kc) {
        if (fullRows) {
            #pragma unroll
            for (int i = 0; i < 16; ++i)
                aReg[i] = X[(rowBase + arow + i * 8) * (long)lda + kc + akk];
        } else {
            #pragma unroll
            for (int i = 0; i < 16; ++i) {
                long gr = rowBase + arow + i * 8;
                aReg[i] = (gr < N) ? X[gr * (long)lda + kc + akk] : 0.f;
            }
        }
        #pragma unroll
        for (int i = 0; i < 16; ++i)
            bReg[i] = W[(long)(kc + bkr0 + i * 2) * dout + colBase + bcc];
    };

    loadChunk(0);
    for (int kc = 0; kc < K; kc += KC) {
        // Spill staged regs to LDS
        #pragma unroll
        for (int i = 0; i < 16; ++i) As[(arow + i * 8) * LPAD + akk] = aReg[i];
        #pragma unroll
        for (int i = 0; i < 16; ++i) Bs[bcc * LPAD + bkr0 + i * 2] = bReg[i];
        __syncthreads();
        // Issue next chunk's global loads; latency hidden under the WMMAs below
        if (kc + KC < K) loadChunk(kc + KC);
        #pragma unroll
        for (int k4 = 0; k4 < KC / 4; ++k4) {
            const int k = k4 * 4;
            v2f a = *(const v2f*)&As[(w * 16 + am) * LPAD + k + ak];
            #pragma unroll
            for (int j = 0; j < 8; ++j) {
                v2f b = *(const v2f*)&Bs[(j * 16 + am) * LPAD + k + ak];
                acc[j] = __builtin_amdgcn_wmma_f32_16x16x4_f32(
                    false, a, false, b, (short)0, acc[j], false, false);
            }
        }
        __syncthreads();
    }

    // Store D: VGPR r -> row M=r (lanes<16) / M=8+r (lanes>=16); N = lane&15
    #pragma unroll
    for (int j = 0; j < 8; ++j) {
        int col = colBase + j * 16 + am;
        float bv = bias[col];
        #pragma unroll
        for (int r = 0; r < 8; ++r) {
            int mrow = (lane < 16) ? r : (r + 8);
            long grow = rowBase + w * 16 + mrow;
            if (grow < N) Y[grow * (long)dout + col] = acc[j][r] + bv;
        }
    }
}

// ---------------------------------------------------------------------------
// Per-column sum & sumsq (batch stats). blockDim.x == C; 512 rows per block.
__global__ void k_stats(const float* __restrict__ Y, int N, int C,
                        float* __restrict__ stats) {
    int c = threadIdx.x;
    long r0 = (long)blockIdx.x * 512;
    long r1 = r0 + 512; if (r1 > N) r1 = N;
    float s = 0.f, s2 = 0.f;
    for (long r = r0; r < r1; ++r) {
        float v = Y[r * C + c];
        s += v; s2 += v * v;
    }
    unsafeAtomicAdd(&stats[c], s);
    unsafeAtomicAdd(&stats[C + c], s2);
}

// BN (batch stats) + ReLU, in place. mode 1: Y = (1+eps)*linmap + relu(bn(Y)),
// linmap col<128 -> attr[r][c], col>=128 -> CS[r/rpi][c-128]
__global__ void k_apply(float* __restrict__ Y, int N, int C,
                        const float* __restrict__ stats,
                        const float* __restrict__ g, const float* __restrict__ be,
                        int mode, const float* __restrict__ eps,
                        const float* __restrict__ attr, const float* __restrict__ CS,
                        int rpi) {
    long i = (long)blockIdx.x * blockDim.x + threadIdx.x;
    if (i >= (long)N * C) return;
    int c = (int)(i % C); long r = i / C;
    float invN = 1.0f / (float)N;
    float mean = stats[c] * invN;
    float var  = stats[C + c] * invN - mean * mean;
    float sc   = g[c] / sqrtf(var + BN_EPS);
    float v = (Y[i] - mean) * sc + be[c];
    v = fmaxf(v, 0.f);
    if (mode == 1) {
        float e = 1.0f + eps[0];
        float cl = (c < 128) ? attr[r * 128 + c] : CS[(r / rpi) * 128 + (c - 128)];
        v = e * cl + v;
    }
    Y[i] = v;
}

// ---------------------------------------------------------------------------
struct Mlp { const float *W1, *b1, *g1, *be1, *W2, *b2, *g2, *be2; };

extern "C" void kernel_launch(void* const* d_in, const int* in_sizes, int n_in,
                              void* d_out, int out_size, void* d_ws, size_t ws_size,
                              hipStream_t stream) {
    const float* edge_attr = (const float*)d_in[0];
    const float* cycle_attr = (const float*)d_in[1];

    // --- Detect params flattening order from leaf sizes ---------------------
    // Top-level (insertion): edge1,edge2,cyc1,cyc2,eps_e1,eps_e2,eps_cycle
    // Top-level (sorted)   : cyc1,cyc2,edge1,edge2,eps_cycle,eps_e1,eps_e2
    bool insertion_top = (in_sizes[2] == 5 * H_DIM * 128);   // edge1.W1 = 81920
    // Inner (insertion): W1,b1,g1,be1,W2,b2,g2,be2 ; (sorted): W1,W2,b1,b2,be1,be2,g1,g2
    bool inner_sorted = (in_sizes[3] > 1024);
    auto F = [&](int i) -> const float* { return (const float*)d_in[i]; };
    auto mk = [&](int b) -> Mlp {
        Mlp m;
        if (inner_sorted) {
            m.W1 = F(b+0); m.W2 = F(b+1); m.b1 = F(b+2); m.b2 = F(b+3);
            m.be1 = F(b+4); m.be2 = F(b+5); m.g1 = F(b+6); m.g2 = F(b+7);
        } else {
            m.W1 = F(b+0); m.b1 = F(b+1); m.g1 = F(b+2); m.be1 = F(b+3);
            m.W2 = F(b+4); m.b2 = F(b+5); m.g2 = F(b+6); m.be2 = F(b+7);
        }
        return m;
    };
    Mlp edge1, edge2, cyc1, cyc2;
    const float *eps_e1, *eps_e2, *eps_cycle;
    if (insertion_top) {
        edge1 = mk(2); edge2 = mk(10); cyc1 = mk(18); cyc2 = mk(26);
        eps_e1 = F(34); eps_e2 = F(35); eps_cycle = F(36);
    } else {
        cyc1 = mk(2); cyc2 = mk(10); edge1 = mk(18); edge2 = mk(26);
        eps_cycle = F(34); eps_e1 = F(35); eps_e2 = F(36);
    }
    const int* e2c1_src = (const int*)d_in[37];
    const int* e2c1_dst = (const int*)d_in[38];
    const int* e2c2_src = (const int*)d_in[39];
    const int* e2c2_dst = (const int*)d_in[40];
    const int* c2e1_src = (const int*)d_in[41];
    const int* c2e1_dst = (const int*)d_in[42];
    const int* c2e2_src = (const int*)d_in[43];
    const int* c2e2_dst = (const int*)d_in[44];

    // --- Workspace layout (floats) -----------------------------------------
    float* ws = (float*)d_ws;
    float* Xc = ws;                               // [RC, 640]
    float* B1 = Xc + (long)RC_ROWS * 640;         // [RC, 128] (reused, RC >= RE)
    float* B2 = B1 + (long)RC_ROWS * 128;         // [RC, 256]
    float* B3 = B2 + (long)RC_ROWS * 256;         // [RC, 128] (lvl)
    float* Xe = B3 + (long)RC_ROWS * 128;         // [RE, 256]
    float* CS = Xe + (long)RE_ROWS * 256;         // [NC, 128]
    float* ST = CS + (long)NC_INST * 128;         // 8 * 512 stats
    float* OutE = (float*)d_out;                  // [RE, 128]
    float* OutC = OutE + (long)RE_ROWS * 128;     // [RC, 128]

    const int T = 256;
    auto cdiv = [](long a, long b) { return (unsigned)((a + b - 1) / b); };

    // Stats zero
    k_zero<<<cdiv(8 * 512, T), T, 0, stream>>>(ST, 8 * 512);

    // --- Cycle-side gathers -------------------------------------------------
    k_init_xc<<<cdiv((long)RC_ROWS * 128, T), T, 0, stream>>>(Xc, cycle_attr);
    // lift_aggr = [e2c2.idp | e2c2.s | e2c1.idp | e2c1.s] ; cols 512.. = cycle_attr
    k_scatter<<<cdiv(M2_CNT, 8), T, 0, stream>>>(edge_attr, e2c2_src, e2c2_dst,
                                                 M2_CNT, Xc, 640, 0, nullptr);
    k_scatter<<<cdiv(M1_CNT, 8), T, 0, stream>>>(edge_attr, e2c1_src, e2c1_dst,
                                                 M1_CNT, Xc, 640, 256, nullptr);
    k_sum_broadcast<<<cdiv((long)NC_INST * 128, T), T, 0, stream>>>(Xc, 640, 0, 128, 6, NC_INST);
    k_sum_broadcast<<<cdiv((long)NC_INST * 128, T), T, 0, stream>>>(Xc, 640, 256, 384, 6, NC_INST);
    k_sum_per_inst<<<cdiv((long)NC_INST * 128, T), T, 0, stream>>>(cycle_attr, CS, 6, NC_INST);

    dim3 gC128(cdiv(RC_ROWS, 128), 1), gC256(cdiv(RC_ROWS, 128), 2);
    dim3 gE128(cdiv(RE_ROWS, 128), 1);
    unsigned sC = cdiv(RC_ROWS, 512), sE = cdiv(RE_ROWS, 512);
    unsigned aC128 = cdiv((long)RC_ROWS * 128, T), aC256 = cdiv((long)RC_ROWS * 256, T);
    unsigned aE128 = cdiv((long)RE_ROWS * 128, T);

    // --- cyc1 MLP: [RC,512] -> 128 -> 256 ----------------------------------
    k_gemm<<<gC128, T, 0, stream>>>(Xc, RC_ROWS, 512, 640, cyc1.W1, 128, cyc1.b1, B1);
    k_stats<<<sC, 128, 0, stream>>>(B1, RC_ROWS, 128, ST + 0 * 512);
    k_apply<<<aC128, T, 0, stream>>>(B1, RC_ROWS, 128, ST + 0 * 512, cyc1.g1, cyc1.be1,
                                     0, nullptr, nullptr, nullptr, 0);
    k_gemm<<<gC256, T, 0, stream>>>(B1, RC_ROWS, 128, 128, cyc1.W2, 256, cyc1.b2, B2);
    k_stats<<<sC, 256, 0, stream>>>(B2, RC_ROWS, 256, ST + 1 * 512);
    // B2 = (1+eps_cycle)*cycle_linmap + relu(bn(B2))
    k_apply<<<aC256, T, 0, stream>>>(B2, RC_ROWS, 256, ST + 1 * 512, cyc1.g2, cyc1.be2,
                                     1, eps_cycle, cycle_attr, CS, 6);

    // --- cyc2 MLP: [RC,256] -> 128 -> 128 -> cycle_out ---------------------
    k_gemm<<<gC128, T, 0, stream>>>(B2, RC_ROWS, 256, 256, cyc2.W1, 128, cyc2.b1, B1);
    k_stats<<<sC, 128, 0, stream>>>(B1, RC_ROWS, 128, ST + 2 * 512);
    k_apply<<<aC128, T, 0, stream>>>(B1, RC_ROWS, 128, ST + 2 * 512, cyc2.g1, cyc2.be1,
                                     0, nullptr, nullptr, nullptr, 0);
    k_gemm<<<gC128, T, 0, stream>>>(B1, RC_ROWS, 128, 128, cyc2.W2, 128, cyc2.b2, OutC);
    k_stats<<<sC, 128, 0, stream>>>(OutC, RC_ROWS, 128, ST + 3 * 512);
    k_apply<<<aC128, T, 0, stream>>>(OutC, RC_ROWS, 128, ST + 3 * 512, cyc2.g2, cyc2.be2,
                                     0, nullptr, nullptr, nullptr, 0);

    // --- edge1 MLP (lvl): [RC,640] -> 128 -> 128 ---------------------------
    k_gemm<<<gC128, T, 0, stream>>>(Xc, RC_ROWS, 640, 640, edge1.W1, 128, edge1.b1, B1);
    k_stats<<<sC, 128, 0, stream>>>(B1, RC_ROWS, 128, ST + 4 * 512);
    k_apply<<<aC128, T, 0, stream>>>(B1, RC_ROWS, 128, ST + 4 * 512, edge1.g1, edge1.be1,
                                     0, nullptr, nullptr, nullptr, 0);
    k_gemm<<<gC128, T, 0, stream>>>(B1, RC_ROWS, 128, 128, edge1.W2, 128, edge1.b2, B3);
    k_stats<<<sC, 128, 0, stream>>>(B3, RC_ROWS, 128, ST + 5 * 512);
    k_apply<<<aC128, T, 0, stream>>>(B3, RC_ROWS, 128, ST + 5 * 512, edge1.g2, edge1.be2,
                                     0, nullptr, nullptr, nullptr, 0);

    // --- Edge-side combine (linear fold of linmap + scaled gathers) --------
    k_init_xe<<<cdiv((long)RE_ROWS * 128, T), T, 0, stream>>>(Xe, edge_attr, eps_e1);
    k_scatter<<<cdiv(M1_CNT, 8), T, 0, stream>>>(B3, c2e1_src, c2e1_dst,
                                                 M1_CNT, Xe, 256, 0, eps_e2);
    k_scatter<<<cdiv(M2_CNT, 8), T, 0, stream>>>(B3, c2e2_src, c2e2_dst,
                                                 M2_CNT, Xe, 256, 0, nullptr);
    k_sum_broadcast<<<cdiv((long)NE_INST * 128, T), T, 0, stream>>>(Xe, 256, 0, 128, 2, NE_INST);

    // --- edge2 MLP: [RE,256] -> 128 -> 128 -> edge_out ---------------------
    k_gemm<<<gE128, T, 0, stream>>>(Xe, RE_ROWS, 256, 256, edge2.W1, 128, edge2.b1, B1);
    k_stats<<<sE, 128, 0, stream>>>(B1, RE_ROWS, 128, ST + 6 * 512);
    k_apply<<<aE128, T, 0, stream>>>(B1, RE_ROWS, 128, ST + 6 * 512, edge2.g1, edge2.be1,
                                     0, nullptr, nullptr, nullptr, 0);
    k_gemm<<<gE128, T, 0, stream>>>(B1, RE_ROWS, 128, 128, edge2.W2, 128, edge2.b2, OutE);
    k_stats<<<sE, 128, 0, stream>>>(OutE, RE_ROWS, 128, ST + 7 * 512);
    k_apply<<<aE128, T, 0, stream>>>(OutE, RE_ROWS, 128, ST + 7 * 512, edge2.g2, edge2.be2,
                                     0, nullptr, nullptr, nullptr, 0);

    (void)in_sizes; (void)n_in; (void)out_size; (void)ws_size;
}